// GATClassifier_81209241633237
// MI455X (gfx1250) — compile-verified
//
#include <hip/hip_runtime.h>
#include <math.h>

typedef __attribute__((ext_vector_type(2))) float v2f;
typedef __attribute__((ext_vector_type(8))) float v8f;

#define NN 100000
#define EE 1600000
#define GG 64
#define HIDDEN 128
#define NHEADS 4

// ---------------- float atomic max via int punning ----------------
__device__ __forceinline__ void atomicMaxF(float* addr, float val) {
  if (val >= 0.0f) atomicMax((int*)addr, __float_as_int(val));
  else             atomicMin((unsigned int*)addr, __float_as_uint(val));
}

// ---------------- weight transpose: Wt[c*128 + k] = W[k*ncols + c]  (K = 128)
__global__ void transpose_w(const float* __restrict__ W, float* __restrict__ Wt, int ncols) {
  const int t = blockIdx.x * blockDim.x + threadIdx.x;
  if (t >= HIDDEN * ncols) return;
  const int k = t / ncols, c = t - k * ncols;
  Wt[(size_t)c * HIDDEN + k] = W[t];
}

// ---------------- WMMA f32 GEMM: C[nrows,ncols] = act(A[nrows,128] @ Wt^T + bias)
// Wt is [ncols,128] (pre-transposed). block = 32*(ncols/16) threads, one wave per
// 16-col tile; grid.x = nrows/16 (nrows MUST be a multiple of 16).
// A tile (16x128, 8KB) staged in LDS once per block.
__global__ void gemm128_wmma(const float* __restrict__ A,
                             const float* __restrict__ Wt,
                             const float* __restrict__ bias,
                             float* __restrict__ C,
                             int ncols, int act) {
  __shared__ float As[16 * HIDDEN];

  const int lane = threadIdx.x & 31;
  const int wave = threadIdx.x >> 5;
  const int rowBase = blockIdx.x * 16;
  const int hi = (lane >= 16) ? 1 : 0;
  const int l15 = lane & 15;
  const int col = wave * 16 + l15;

  // cooperative stage of contiguous 16x128 A tile
  {
    const float4* src = (const float4*)(A + (size_t)rowBase * HIDDEN);
    float4* dstl = (float4*)As;
    for (int i = threadIdx.x; i < (16 * HIDDEN) / 4; i += blockDim.x) dstl[i] = src[i];
  }
  __syncthreads();

  v8f acc;
  const float bv = bias ? bias[col] : 0.0f;
#pragma unroll
  for (int j = 0; j < 8; ++j) acc[j] = bv;

  const float* __restrict__ brow = Wt + (size_t)col * HIDDEN;  // contiguous K per column

#pragma unroll 8
  for (int k = 0; k < HIDDEN; k += 4) {
    const int ka = k + hi * 2;                   // K index of VGPR j=0 for this lane half
    v2f a = *(const v2f*)&As[l15 * HIDDEN + ka]; // ds_load_b64
    v2f b = *(const v2f*)&brow[ka];              // global_load_b64 (contiguous K)
    acc = __builtin_amdgcn_wmma_f32_16x16x4_f32(false, a, false, b, (short)0, acc,
                                                false, false);
  }

#pragma unroll
  for (int j = 0; j < 8; ++j) {
    float v = acc[j];
    if (act == 1) v = fmaxf(v, 0.0f);
    C[(size_t)(rowBase + j + hi * 8) * ncols + col] = v;   // unguarded: nrows % 16 == 0
  }
}

// ---------------- per-(node,head) attention scores; also inits m=-inf, denom=0
__global__ void attn_scores(const float* __restrict__ xh,
                            const float* __restrict__ a_src,
                            const float* __restrict__ a_dst,
                            float* __restrict__ s_src,
                            float* __restrict__ s_dst,
                            float* __restrict__ m,
                            float* __restrict__ denom,
                            int n, int heads) {
  const int t = blockIdx.x * blockDim.x + threadIdx.x;
  if (t >= n * heads) return;
  const int node = t / heads, h = t - node * heads;
  const int C = HIDDEN / heads;
  const float* xr = xh + (size_t)node * HIDDEN + h * C;
  const float* as = a_src + h * C;
  const float* ad = a_dst + h * C;
  float ss = 0.f, sd = 0.f;
  for (int c = 0; c < C; ++c) { const float v = xr[c]; ss += v * as[c]; sd += v * ad[c]; }
  s_src[t] = ss;
  s_dst[t] = sd;
  m[t] = -INFINITY;
  denom[t] = 0.f;
}

// ---------------- pass 1: segment max of leaky_relu scores over incoming edges (+self loops)
__global__ void edge_max(const int* __restrict__ src,
                         const int* __restrict__ dst,
                         const float* __restrict__ s_src,
                         const float* __restrict__ s_dst,
                         float* __restrict__ m,
                         int E, int n, int heads) {
  const int total = (E + n) * heads;
  const int t = blockIdx.x * blockDim.x + threadIdx.x;
  if (t >= total) return;
  const int e = t / heads, h = t - e * heads;
  int s, d;
  if (e < E) { s = src[e]; d = dst[e]; } else { s = d = e - E; }
  float v = s_src[s * heads + h] + s_dst[d * heads + h];
  v = (v > 0.f) ? v : 0.2f * v;
  atomicMaxF(&m[d * heads + h], v);
}

// ---------------- pass 2: one wave per edge; accumulate exp * xh[src] and denom
__global__ void edge_accum(const int* __restrict__ src,
                           const int* __restrict__ dst,
                           const float* __restrict__ s_src,
                           const float* __restrict__ s_dst,
                           const float* __restrict__ m,
                           const float* __restrict__ xh,
                           float* __restrict__ numer,
                           float* __restrict__ denom,
                           int E, int n, int heads) {
  const int lane = threadIdx.x & 31;
  const int e = blockIdx.x * (blockDim.x >> 5) + (threadIdx.x >> 5);
  if (e >= E + n) return;
  int s, d;
  if (e < E) { s = src[e]; d = dst[e]; } else { s = d = e - E; }
  const int C = HIDDEN / heads;
  const int c0 = lane * 4;
  const int h = c0 / C;
  float v = s_src[s * heads + h] + s_dst[d * heads + h];
  v = (v > 0.f) ? v : 0.2f * v;
  const float ex = expf(v - m[d * heads + h]);
  const float4 xv = *(const float4*)(xh + (size_t)s * HIDDEN + c0);   // global_load_b128
  float* np_ = numer + (size_t)d * HIDDEN + c0;
  atomicAdd(np_ + 0, ex * xv.x);
  atomicAdd(np_ + 1, ex * xv.y);
  atomicAdd(np_ + 2, ex * xv.z);
  atomicAdd(np_ + 3, ex * xv.w);
  if ((c0 % C) == 0) atomicAdd(&denom[d * heads + h], ex);
}

// ---------------- per-node: divide, +bias, LayerNorm, ELU, residual (one wave per node)
__global__ void gat_finalize(const float* __restrict__ numer,
                             const float* __restrict__ denom,
                             const float* __restrict__ bias,
                             const float* __restrict__ ln_g,
                             const float* __restrict__ ln_b,
                             float* __restrict__ h,     // in: residual, out: new h
                             int n, int heads) {
  const int lane = threadIdx.x & 31;
  const int node = blockIdx.x * (blockDim.x >> 5) + (threadIdx.x >> 5);
  if (node >= n) return;
  const int C = HIDDEN / heads;
  const int c0 = lane * 4;
  float vals[4], res[4];
  float lsum = 0.f;
#pragma unroll
  for (int j = 0; j < 4; ++j) {
    const int c = c0 + j;
    const int hd = c / C;
    const float dn = denom[node * heads + hd] + 1e-16f;
    const float v = numer[(size_t)node * HIDDEN + c] / dn + bias[c];
    res[j] = h[(size_t)node * HIDDEN + c];
    vals[j] = v;
    lsum += v;
  }
  for (int off = 16; off >= 1; off >>= 1) lsum += __shfl_xor(lsum, off);
  const float mu = lsum / (float)HIDDEN;
  float lvar = 0.f;
#pragma unroll
  for (int j = 0; j < 4; ++j) { const float dd = vals[j] - mu; lvar += dd * dd; }
  for (int off = 16; off >= 1; off >>= 1) lvar += __shfl_xor(lvar, off);
  const float rstd = rsqrtf(lvar / (float)HIDDEN + 1e-5f);
#pragma unroll
  for (int j = 0; j < 4; ++j) {
    const int c = c0 + j;
    float y = (vals[j] - mu) * rstd * ln_g[c] + ln_b[c];
    y = (y > 0.f) ? y : (expf(y) - 1.0f);               // ELU
    h[(size_t)node * HIDDEN + c] = y + res[j];
  }
}

// ---------------- gate head second layer: gate[n] = gtmp[n,:64] . w2 + b2
__global__ void gate_dot(const float* __restrict__ gt, const float* __restrict__ w2,
                         const float* __restrict__ b2, float* __restrict__ gate, int n) {
  const int t = blockIdx.x * blockDim.x + threadIdx.x;
  if (t >= n) return;
  const float* r = gt + (size_t)t * 64;
  float s = b2[0];
#pragma unroll 8
  for (int j = 0; j < 64; ++j) s += r[j] * w2[j];
  gate[t] = s;
}

__global__ void pool_init(float* gm, float* gs, int G) {
  const int t = threadIdx.x;
  if (t < G) { gm[t] = -INFINITY; gs[t] = 0.f; }
}

__global__ void pool_max(const float* __restrict__ gate, const int* __restrict__ batch,
                         float* __restrict__ gm, int n) {
  const int t = blockIdx.x * blockDim.x + threadIdx.x;
  if (t >= n) return;
  atomicMaxF(&gm[batch[t]], gate[t]);
}

// one wave per node: accumulate exp(gate-gm)*h into pooled, and gs
__global__ void pool_accum(const float* __restrict__ gate, const int* __restrict__ batch,
                           const float* __restrict__ gm, const float* __restrict__ h,
                           float* __restrict__ gs, float* __restrict__ pooled, int n) {
  const int lane = threadIdx.x & 31;
  const int node = blockIdx.x * (blockDim.x >> 5) + (threadIdx.x >> 5);
  if (node >= n) return;
  const int b = batch[node];
  const float ge = expf(gate[node] - gm[b]);
  if (lane == 0) atomicAdd(&gs[b], ge);
  const int c0 = lane * 4;
  const float4 hv = *(const float4*)(h + (size_t)node * HIDDEN + c0);
  float* p = pooled + (size_t)b * HIDDEN + c0;
  atomicAdd(p + 0, ge * hv.x);
  atomicAdd(p + 1, ge * hv.y);
  atomicAdd(p + 2, ge * hv.z);
  atomicAdd(p + 3, ge * hv.w);
}

// final classifier: thread per graph
__global__ void classifier(const float* __restrict__ pooled, const float* __restrict__ gs,
                           const float* __restrict__ w1, const float* __restrict__ b1,
                           const float* __restrict__ w2, const float* __restrict__ b2,
                           float* __restrict__ out, int G) {
  const int g = blockIdx.x * blockDim.x + threadIdx.x;
  if (g >= G) return;
  const float inv = 1.0f / (gs[g] + 1e-16f);
  float o[10];
#pragma unroll
  for (int k = 0; k < 10; ++k) o[k] = b2[k];
  for (int j = 0; j < 64; ++j) {
    float s = b1[j];
    for (int c = 0; c < 128; ++c) s += pooled[g * 128 + c] * inv * w1[c * 64 + j];
    s = fmaxf(s, 0.f);
#pragma unroll
    for (int k = 0; k < 10; ++k) o[k] += s * w2[j * 10 + k];
  }
  for (int k = 0; k < 10; ++k) out[g * 10 + k] = o[k];
}

extern "C" void kernel_launch(void* const* d_in, const int* in_sizes, int n_in,
                              void* d_out, int out_size, void* d_ws, size_t ws_size,
                              hipStream_t stream) {
  const float* x     = (const float*)d_in[0];
  const int*   ei    = (const int*)d_in[1];          // [2,E]
  const int*   batch = (const int*)d_in[3];
  const float* np_w  = (const float*)d_in[4];
  const float* np_b  = (const float*)d_in[5];
  const float* gw[3] = {(const float*)d_in[8],  (const float*)d_in[12], (const float*)d_in[16]};
  const float* gas[3]= {(const float*)d_in[9],  (const float*)d_in[13], (const float*)d_in[17]};
  const float* gad[3]= {(const float*)d_in[10], (const float*)d_in[14], (const float*)d_in[18]};
  const float* gb[3] = {(const float*)d_in[11], (const float*)d_in[15], (const float*)d_in[19]};
  const float* lng[3]= {(const float*)d_in[20], (const float*)d_in[22], (const float*)d_in[24]};
  const float* lnb[3]= {(const float*)d_in[21], (const float*)d_in[23], (const float*)d_in[25]};
  const float* gate_w1 = (const float*)d_in[26];
  const float* gate_b1 = (const float*)d_in[27];
  const float* gate_w2 = (const float*)d_in[28];
  const float* gate_b2 = (const float*)d_in[29];
  const float* cls_w1  = (const float*)d_in[30];
  const float* cls_b1  = (const float*)d_in[31];
  const float* cls_w2  = (const float*)d_in[32];
  const float* cls_b2  = (const float*)d_in[33];

  const int* src = ei;
  const int* dst = ei + EE;

  // workspace layout (floats)
  float* ws = (float*)d_ws;
  size_t off = 0;
  float* h     = ws + off; off += (size_t)NN * HIDDEN;
  float* xh    = ws + off; off += (size_t)NN * HIDDEN;
  float* numer = ws + off; off += (size_t)NN * HIDDEN;
  float* s_src = ws + off; off += (size_t)NN * NHEADS;
  float* s_dst = ws + off; off += (size_t)NN * NHEADS;
  float* m     = ws + off; off += (size_t)NN * NHEADS;
  float* denom = ws + off; off += (size_t)NN * NHEADS;
  float* gate  = ws + off; off += (size_t)NN;
  float* gm    = ws + off; off += GG;
  float* gs    = ws + off; off += GG;
  float* pooled= ws + off; off += (size_t)GG * HIDDEN;
  float* wt_np = ws + off; off += (size_t)HIDDEN * HIDDEN;
  float* wt_g0 = ws + off; off += (size_t)HIDDEN * HIDDEN;
  float* wt_g1 = ws + off; off += (size_t)HIDDEN * HIDDEN;
  float* wt_g2 = ws + off; off += (size_t)HIDDEN * HIDDEN;
  float* wt_gate=ws + off; off += (size_t)HIDDEN * 64;
  float* wt_g[3] = {wt_g0, wt_g1, wt_g2};
  float* gtmp  = xh;                                    // reuse xh after GAT layers

  const int rowBlocks = NN / 16;                        // 100000 % 16 == 0

  // transpose all K=128 weight matrices once (deterministic, every call)
  transpose_w<<<(HIDDEN * HIDDEN + 255) / 256, 256, 0, stream>>>(np_w, wt_np, HIDDEN);
  for (int l = 0; l < 3; ++l)
    transpose_w<<<(HIDDEN * HIDDEN + 255) / 256, 256, 0, stream>>>(gw[l], wt_g[l], HIDDEN);
  transpose_w<<<(HIDDEN * 64 + 255) / 256, 256, 0, stream>>>(gate_w1, wt_gate, 64);

  // h = relu(x @ np_w + np_b)
  gemm128_wmma<<<rowBlocks, 256, 0, stream>>>(x, wt_np, np_b, h, HIDDEN, 1);

  for (int l = 0; l < 3; ++l) {
    const int heads = (l == 2) ? 1 : NHEADS;
    // xh = h @ W_l
    gemm128_wmma<<<rowBlocks, 256, 0, stream>>>(h, wt_g[l], nullptr, xh, HIDDEN, 0);
    // per-node per-head projected attention scores; inits m/denom
    {
      const int tot = NN * heads;
      attn_scores<<<(tot + 255) / 256, 256, 0, stream>>>(xh, gas[l], gad[l],
                                                         s_src, s_dst, m, denom, NN, heads);
    }
    hipMemsetAsync(numer, 0, (size_t)NN * HIDDEN * sizeof(float), stream);
    {
      const int tot = (EE + NN) * heads;
      edge_max<<<(tot + 255) / 256, 256, 0, stream>>>(src, dst, s_src, s_dst, m, EE, NN, heads);
    }
    {
      const int waves = EE + NN;               // one wave per edge (incl. self loops)
      edge_accum<<<(waves + 7) / 8, 256, 0, stream>>>(src, dst, s_src, s_dst, m, xh,
                                                      numer, denom, EE, NN, heads);
    }
    gat_finalize<<<(NN + 7) / 8, 256, 0, stream>>>(numer, denom, gb[l], lng[l], lnb[l],
                                                   h, NN, heads);
  }

  // gate = relu(h@gate_w1+b1) @ gate_w2 + b2
  gemm128_wmma<<<rowBlocks, 128, 0, stream>>>(h, wt_gate, gate_b1, gtmp, 64, 1);
  gate_dot<<<(NN + 255) / 256, 256, 0, stream>>>(gtmp, gate_w2, gate_b2, gate, NN);

  pool_init<<<1, 64, 0, stream>>>(gm, gs, GG);
  hipMemsetAsync(pooled, 0, (size_t)GG * HIDDEN * sizeof(float), stream);
  pool_max<<<(NN + 255) / 256, 256, 0, stream>>>(gate, batch, gm, NN);
  pool_accum<<<(NN + 7) / 8, 256, 0, stream>>>(gate, batch, gm, h, gs, pooled, NN);

  classifier<<<1, 64, 0, stream>>>(pooled, gs, cls_w1, cls_b1, cls_w2, cls_b2,
                                   (float*)d_out, GG);
}